// neg_sampling_loss_36945308680351
// MI455X (gfx1250) — compile-verified
//
#include <hip/hip_runtime.h>
#include <math.h>

// Problem constants (reference: N=50000, D=256, K_POS=32, K_NEG=20)
#define D_DIM   256
#define KPOS    32
#define KNEG    20
#define SLOTS   64                 // 52 real gather slots padded to 4 WMMA tiles of 16
#define WAVES_PER_BLOCK 8
#define BLOCK_THREADS   (WAVES_PER_BLOCK * 32)
#define EPS_F   1e-15f

typedef float v2f __attribute__((ext_vector_type(2)));
typedef float v8f __attribute__((ext_vector_type(8)));

// ---------------------------------------------------------------------------
// Main kernel: one wave (wave32) per embedding row n.
//  Phase 1: stage query row emb[n][0..255] into LDS.
//  Phase 2: 4 tiles x 16 gathered rows; dots via V_WMMA_F32_16X16X4_F32,
//           accumulating K=256 in 64 steps of K=4 (full fp32 precision).
//  Phase 3: loss terms + wave reduction + global atomic accumulation.
// ---------------------------------------------------------------------------
__global__ __launch_bounds__(BLOCK_THREADS)
void nsl_main(const float* __restrict__ emb,
              const int*   __restrict__ nbr,       // [N, KPOS]
              const int*   __restrict__ nbr_mask,  // [N, KPOS] 0/1
              const int*   __restrict__ negs,      // [N, KNEG]
              const int*   __restrict__ rmask,     // [N] 0/1
              float*       __restrict__ acc,       // acc[0]=numerator, acc[1]=denominator
              int N)
{
    __shared__ float q[WAVES_PER_BLOCK][D_DIM];    // query rows (1KB/wave)
    __shared__ float dots[WAVES_PER_BLOCK][SLOTS]; // per-wave dot results

    const int lane = threadIdx.x & 31;
    const int wave = threadIdx.x >> 5;
    const int row  = blockIdx.x * WAVES_PER_BLOCK + wave;
    const bool active = (row < N);
    const int  n   = active ? row : 0;             // clamp: inactive waves still run WMMA

    // ---- Phase 1: stage query row into LDS (8 floats/lane as two b128) ----
    {
        const float4* src = (const float4*)(emb + (size_t)n * D_DIM);
        float4 u0 = src[lane * 2 + 0];
        float4 u1 = src[lane * 2 + 1];
        float4* dst = (float4*)&q[wave][0];
        dst[lane * 2 + 0] = u0;
        dst[lane * 2 + 1] = u1;
    }
    __syncthreads();

    // Per-lane K-chunk offset shared by the A (16x4 f32) and B (4x16 f32)
    // layouts: lanes 0-15 hold K={0,1}, lanes 16-31 hold K={2,3}.
    const int koff = (lane < 16) ? 0 : 2;
    const int m16  = lane & 15;                    // this lane's A-matrix row (neighbor)
    const float* qbase = &q[wave][koff];

    #pragma unroll
    for (int t = 0; t < 4; ++t) {
        // Gather index for this lane's neighbor row (slots 52..63 -> dummy row 0)
        int s = t * 16 + m16;
        int g;
        if (s < KPOS)             g = nbr [(size_t)n * KPOS + s];
        else if (s < KPOS + KNEG) g = negs[(size_t)n * KNEG + (s - KPOS)];
        else                      g = 0;
        const float* arow = emb + (size_t)g * D_DIM + koff;

        v8f dacc = {};                             // 16x16 f32 accumulator (8 VGPRs)
        #pragma unroll 4
        for (int k0 = 0; k0 < D_DIM; k0 += 4) {
            v2f a = *(const v2f*)(arow  + k0);     // global_load_b64 (L2 gather)
            v2f b = *(const v2f*)(qbase + k0);     // ds_load_b64 (query broadcast)
            dacc = __builtin_amdgcn_wmma_f32_16x16x4_f32(
                       /*neg_a=*/false, a, /*neg_b=*/false, b,
                       /*c_mod=*/(short)0, dacc, /*reuse_a=*/false, /*reuse_b=*/false);
        }
        // D layout: VGPR r -> M=r (lanes 0-15) / M=r+8 (lanes 16-31); all N columns
        // are identical (B columns were replicated), so lanes 0 and 16 hold all dots.
        if (lane == 0) {
            #pragma unroll
            for (int r = 0; r < 8; ++r) dots[wave][t * 16 + r] = dacc[r];
        } else if (lane == 16) {
            #pragma unroll
            for (int r = 0; r < 8; ++r) dots[wave][t * 16 + 8 + r] = dacc[r];
        }
    }
    __syncthreads();

    // ---- Phase 3: loss terms (slot = lane for pos, 32+lane for neg) ----
    float dp = dots[wave][lane];
    float mp = (float)nbr_mask[(size_t)n * KPOS + lane];
    float sp = 1.0f / (1.0f + expf(-dp));
    float pos_sum = -logf(sp + EPS_F) * mp;
    float cntf    = mp;
    float neg_sum = 0.0f;
    if (lane < KNEG) {
        float dn = dots[wave][KPOS + lane];
        float sn = 1.0f / (1.0f + expf(-dn));
        neg_sum = -logf(1.0f - sn + EPS_F);
    }

    // wave32 butterfly reduction
    #pragma unroll
    for (int off = 16; off > 0; off >>= 1) {
        pos_sum += __shfl_xor(pos_sum, off, 32);
        cntf    += __shfl_xor(cntf,    off, 32);
        neg_sum += __shfl_xor(neg_sum, off, 32);
    }

    if (active && lane == 0) {
        // cnt==0 rows: reference produces NaN -> replaced by 0 contribution
        float row_loss = (cntf > 0.0f)
                       ? (pos_sum / cntf + neg_sum * (1.0f / (float)KNEG))
                       : 0.0f;
        float rm = (float)rmask[n];
        atomicAdd(&acc[0], row_loss * rm);
        atomicAdd(&acc[1], rm);
    }
}

__global__ void nsl_init(float* acc)
{
    if (threadIdx.x < 2) acc[threadIdx.x] = 0.0f;
}

__global__ void nsl_fin(const float* __restrict__ acc, float* __restrict__ out)
{
    if (threadIdx.x == 0) out[0] = acc[0] / acc[1];
}

extern "C" void kernel_launch(void* const* d_in, const int* in_sizes, int n_in,
                              void* d_out, int out_size, void* d_ws, size_t ws_size,
                              hipStream_t stream)
{
    const float* emb   = (const float*)d_in[0];
    const int*   nbr   = (const int*)  d_in[1];
    const int*   nmask = (const int*)  d_in[2];
    const int*   negs  = (const int*)  d_in[3];
    const int*   rmask = (const int*)  d_in[4];
    const int    N     = in_sizes[4];

    float* acc = (float*)d_ws;

    nsl_init<<<1, 32, 0, stream>>>(acc);

    const int blocks = (N + WAVES_PER_BLOCK - 1) / WAVES_PER_BLOCK;
    nsl_main<<<blocks, BLOCK_THREADS, 0, stream>>>(emb, nbr, nmask, negs, rmask, acc, N);

    nsl_fin<<<1, 32, 0, stream>>>(acc, (float*)d_out);
}